// PulseTrain_7335804141992
// MI455X (gfx1250) — compile-verified
//
#include <hip/hip_runtime.h>
#include <math.h>

// Problem constants (from reference)
#define BROWS 32
#define TLEN  524288
// Tiling
#define TILE   8192                 // elements per block-tile
#define TPB    256                  // 8 waves of 32
#define SEQ    (TILE / TPB)         // 32 elements per thread
#define NTILES (TLEN / TILE)        // 64 tiles per row

typedef __attribute__((ext_vector_type(2))) float v2f;
typedef __attribute__((ext_vector_type(8))) float v8f;

// ---------------------------------------------------------------------------
// Pass 1: per-tile partial sums.  grid = (NTILES, BROWS), block = TPB.
// ---------------------------------------------------------------------------
__global__ void pt_partials(const float* __restrict__ phase,
                            float* __restrict__ partial) {
    const int tile = blockIdx.x;
    const int row  = blockIdx.y;
    const float* p = phase + (size_t)row * TLEN + (size_t)tile * TILE
                           + (size_t)threadIdx.x * SEQ;
    float s = 0.f;
#pragma unroll
    for (int i = 0; i < SEQ; i += 4) {
        float4 v = *(const float4*)(p + i);      // global_load_b128
        s += v.x + v.y + v.z + v.w;
    }
    __shared__ float red[TPB];
    red[threadIdx.x] = s;
    __syncthreads();
    for (int off = TPB / 2; off > 0; off >>= 1) {
        if (threadIdx.x < off) red[threadIdx.x] += red[threadIdx.x + off];
        __syncthreads();
    }
    if (threadIdx.x == 0) partial[row * NTILES + tile] = red[0];
}

// ---------------------------------------------------------------------------
// Pass 2: exclusive scan of the 64 tile partials per row + phase offset.
// grid = BROWS blocks; trivially small (64 sequential adds per row).
// ---------------------------------------------------------------------------
__global__ void pt_scan_partials(const float* __restrict__ partial,
                                 const float* __restrict__ offset,
                                 float* __restrict__ tileBase) {
    const int row = blockIdx.x;
    if (threadIdx.x == 0) {
        float run = offset[row];
        for (int t = 0; t < NTILES; ++t) {
            tileBase[row * NTILES + t] = run;
            run += partial[row * NTILES + t];
        }
    }
}

// ---------------------------------------------------------------------------
// Pass 3: re-read tile (L2-resident), WMMA block scan of thread totals,
// running sum + wrap detect, emit pulses.
// ---------------------------------------------------------------------------
__global__ void pt_emit(const float* __restrict__ phase,
                        const float* __restrict__ tileBase,
                        float* __restrict__ out) {
    const int tile = blockIdx.x;
    const int row  = blockIdx.y;
    const int tid  = threadIdx.x;
    const size_t base = (size_t)row * TLEN + (size_t)tile * TILE;
    const float* p = phase + base + (size_t)tid * SEQ;

    // Load this thread's 32 contiguous values (hits L2 after pass 1).
    float x[SEQ];
#pragma unroll
    for (int i = 0; i < SEQ; i += 4) {
        float4 v = *(const float4*)(p + i);
        x[i + 0] = v.x; x[i + 1] = v.y; x[i + 2] = v.z; x[i + 3] = v.w;
    }
    float total = 0.f;
#pragma unroll
    for (int i = 0; i < SEQ; ++i) total += x[i];

    // ---- block-wide inclusive scan of 256 thread totals via WMMA ----
    // Value v = n*16 + k lives at matrix element (row k, col n) of X.
    __shared__ float totals[TPB];
    __shared__ float coltot[16];
    __shared__ float incl[TPB];
    totals[tid] = total;
    __syncthreads();

    if (tid < 32) {                 // wave 0, EXEC all ones for WMMA
        const int lane = tid;
        const int n    = lane & 15; // column (B/C/D lane dim), also A row M
        const int half = lane >> 4; // selects K pair per ISA VGPR layout
        const int k0   = 2 * half;
        const int k1   = 2 * half + 1;

        // colcum = Lincl(16x16 lower-tri ones) * X  -- 4 chained K=4 WMMAs
        v8f d = {0.f, 0.f, 0.f, 0.f, 0.f, 0.f, 0.f, 0.f};
#pragma unroll
        for (int c = 0; c < 4; ++c) {
            v2f a, b;
            a[0] = ((4 * c + k0) <= n) ? 1.f : 0.f;   // Lincl[m][k], m = lane&15
            a[1] = ((4 * c + k1) <= n) ? 1.f : 0.f;
            b[0] = totals[n * 16 + 4 * c + k0];       // X[k][n]
            b[1] = totals[n * 16 + 4 * c + k1];
            d = __builtin_amdgcn_wmma_f32_16x16x4_f32(
                    false, a, false, b, (short)0, d, false, false);
        }
        // column totals = row 15 of colcum -> lanes with half==1, VGPR 7
        if (half == 1) coltot[n] = d[7];

        // E[m][n] = sum_{k<n} coltot[k]  (exclusive cross-column prefix,
        // coltot fed along K; U = strict-upper-triangular ones)
        v8f e = {0.f, 0.f, 0.f, 0.f, 0.f, 0.f, 0.f, 0.f};
#pragma unroll
        for (int c = 0; c < 4; ++c) {
            v2f a, b;
            a[0] = coltot[4 * c + k0];
            a[1] = coltot[4 * c + k1];
            b[0] = ((4 * c + k0) < n) ? 1.f : 0.f;
            b[1] = ((4 * c + k1) < n) ? 1.f : 0.f;
            e = __builtin_amdgcn_wmma_f32_16x16x4_f32(
                    false, a, false, b, (short)0, e, false, false);
        }
        // inclusive prefix of value n*16 + m,  m = v + 8*half (C/D layout)
#pragma unroll
        for (int v = 0; v < 8; ++v)
            incl[n * 16 + 8 * half + v] = d[v] + e[v];
    }
    __syncthreads();

    // ---- running sum + wrap detection ----
    const float excl = incl[tid] - total;                 // exclusive prefix
    float run = tileBase[row * NTILES + tile] + excl;     // S_{t-1} at entry
    float prevFloor = floorf(run);
    float* o = out + base + (size_t)tid * SEQ;
    const int pos0 = tile * TILE + tid * SEQ;             // position in row

#pragma unroll
    for (int i = 0; i < SEQ; i += 4) {
        float4 v;
        float r[4];
#pragma unroll
        for (int j = 0; j < 4; ++j) {
            run += x[i + j];
            const float f = floorf(run);
            float val = (f > prevFloor) ? rsqrtf(x[i + j]) : 0.f;
            if (pos0 + i + j == 0) val = 0.f;             // out[:,0] == 0
            r[j] = val;
            prevFloor = f;
        }
        v.x = r[0]; v.y = r[1]; v.z = r[2]; v.w = r[3];
        *(float4*)(o + i) = v;                            // global_store_b128
    }
}

// ---------------------------------------------------------------------------
extern "C" void kernel_launch(void* const* d_in, const int* in_sizes, int n_in,
                              void* d_out, int out_size, void* d_ws, size_t ws_size,
                              hipStream_t stream) {
    const float* phase  = (const float*)d_in[0];   // [B, T] f32
    const float* offset = (const float*)d_in[1];   // [B, 1] f32
    float* out = (float*)d_out;                    // [B, T] f32

    float* partial  = (float*)d_ws;                // [B, NTILES]
    float* tileBase = partial + BROWS * NTILES;    // [B, NTILES]

    dim3 grid(NTILES, BROWS);
    pt_partials<<<grid, TPB, 0, stream>>>(phase, partial);
    pt_scan_partials<<<BROWS, 32, 0, stream>>>(partial, offset, tileBase);
    pt_emit<<<grid, TPB, 0, stream>>>(phase, tileBase, out);
}